// BPKDV2_24979529793864
// MI455X (gfx1250) — compile-verified
//
#include <hip/hip_runtime.h>
#include <hip/hip_bf16.h>

// BPKDV2 body/edge KD loss for MI455X (gfx1250, wave32).
// - Single streaming pass over preds_S/preds_T channels 1..13 via
//   global_load_async_to_lds_b128 (ASYNCcnt-pipelined, 8 in flight per wave),
//   NT temporal hint so the 218 MB stream doesn't evict L2-resident labels.
// - Closed-form KL from 5 masked sums per (plane, edge|body).
// - Finalize: per-batch class sums via chained V_WMMA_F32_16X16X4_F32.

#define HW  262144   // 512*512
#define WID 512
#define NPLANE 104   // 8 batches * 13 classes
#define ACCN (NPLANE * 10)

typedef __attribute__((ext_vector_type(2))) float v2f;
typedef __attribute__((ext_vector_type(8))) float v8f;

__device__ __forceinline__ void async_ld_b128_nt(unsigned lds_addr, const void* gaddr) {
  asm volatile("global_load_async_to_lds_b128 %0, %1, off th:TH_LOAD_NT"
               :: "v"(lds_addr), "v"((unsigned long long)gaddr)
               : "memory");
}

__global__ void BPKDV2_zero_kernel(float* __restrict__ acc, int n) {
  int i = blockIdx.x * blockDim.x + threadIdx.x;
  if (i < n) acc[i] = 0.0f;
}

__global__ __launch_bounds__(256) void BPKDV2_main_kernel(
    const float* __restrict__ S, const float* __restrict__ T,
    const int* __restrict__ L, float* __restrict__ acc)
{
  __shared__ __align__(16) float ldsS[8][4][128];  // [wave][chunk][float]
  __shared__ __align__(16) float ldsT[8][4][128];
  __shared__ float wgAcc[10];

  const int tid  = threadIdx.x;
  const int wave = tid >> 5;
  const int lane = tid & 31;
  const int wg   = blockIdx.x;
  const int plane = wg >> 6;        // 0..103
  const int tile  = wg & 63;        // 0..63  (8 rows per tile)
  const int b = plane / 13;
  const int c = plane % 13 + 1;     // class id == preds channel index
  const int row = tile * 8 + wave;  // 0..511 (wave-uniform)

  const size_t planeOff = ((size_t)b * 14 + (size_t)c) * HW;
  const float* Srow = S + planeOff + (size_t)row * WID;
  const float* Trow = T + planeOff + (size_t)row * WID;
  const int*   Lrow = L + (size_t)b * HW + (size_t)row * WID;

  if (tid < 10) wgAcc[tid] = 0.0f;

  // ---- issue 8 async global->LDS B128 copies (4 chunks x {S,T}), NT ----
#pragma unroll
  for (int k = 0; k < 4; ++k) {
    unsigned aS = (unsigned)(size_t)&ldsS[wave][k][lane * 4];
    unsigned aT = (unsigned)(size_t)&ldsT[wave][k][lane * 4];
    async_ld_b128_nt(aS, Srow + k * 128 + lane * 4);
    async_ld_b128_nt(aT, Trow + k * 128 + lane * 4);
  }

  float cntE = 0.f, AE = 0.f, BE = 0.f, CE = 0.f, DE = 0.f;
  float cntB = 0.f, AB = 0.f, BB = 0.f, CB = 0.f, DB = 0.f;

#pragma unroll
  for (int k = 0; k < 4; ++k) {
    // in-order async-load completion: chunk k done when <= 2*(3-k) remain
    switch (k) {
      case 0:  asm volatile("s_wait_asynccnt 0x6" ::: "memory"); break;
      case 1:  asm volatile("s_wait_asynccnt 0x4" ::: "memory"); break;
      case 2:  asm volatile("s_wait_asynccnt 0x2" ::: "memory"); break;
      default: asm volatile("s_wait_asynccnt 0x0" ::: "memory"); break;
    }
    const int x0 = k * 128 + lane * 4;
    const float4 s4 = *(const float4*)&ldsS[wave][k][lane * 4];
    const float4 t4 = *(const float4*)&ldsT[wave][k][lane * 4];

    // labels: cross neighborhood; OOB pads behave as class 0 (never == c)
    const int4 lc = *(const int4*)&Lrow[x0];
    const int4 lu = (row > 0)   ? *(const int4*)&Lrow[x0 - WID] : make_int4(0, 0, 0, 0);
    const int4 ld = (row < 511) ? *(const int4*)&Lrow[x0 + WID] : make_int4(0, 0, 0, 0);
    const int ll = (x0 > 0)   ? Lrow[x0 - 1] : 0;
    const int lr = (x0 < 508) ? Lrow[x0 + 4] : 0;

    const int mc[4] = { lc.x == c, lc.y == c, lc.z == c, lc.w == c };
    const int mu[4] = { lu.x == c, lu.y == c, lu.z == c, lu.w == c };
    const int md[4] = { ld.x == c, ld.y == c, ld.z == c, ld.w == c };
    const int mL[4] = { ll == c, mc[0], mc[1], mc[2] };
    const int mR[4] = { mc[1], mc[2], mc[3], lr == c };
    const float sv[4] = { s4.x, s4.y, s4.z, s4.w };
    const float tv[4] = { t4.x, t4.y, t4.z, t4.w };

#pragma unroll
    for (int j = 0; j < 4; ++j) {
      const int dil = mc[j] | mu[j] | md[j] | mL[j] | mR[j];
      const int ero = mc[j] & mu[j] & md[j] & mL[j] & mR[j];
      if (dil) {                       // pixel contributes (edge or body)
        const float eT = __expf(tv[j]);
        const float eS = __expf(sv[j]);
        if (ero) {                     // body = erode
          cntB += 1.f; AB += eT; BB += eT * tv[j]; CB += eT * sv[j]; DB += eS;
        } else {                       // edge = dilate & !erode
          cntE += 1.f; AE += eT; BE += eT * tv[j]; CE += eT * sv[j]; DE += eS;
        }
      }
    }
  }

  // ---- wave32 shuffle reduction, then LDS, then one global atomic set ----
  float v[10] = { cntE, AE, BE, CE, DE, cntB, AB, BB, CB, DB };
#pragma unroll
  for (int i = 0; i < 10; ++i) {
#pragma unroll
    for (int off = 16; off > 0; off >>= 1)
      v[i] += __shfl_down(v[i], off, 32);
  }

  __syncthreads();
  if (lane == 0) {
#pragma unroll
    for (int i = 0; i < 10; ++i) atomicAdd(&wgAcc[i], v[i]);
  }
  __syncthreads();
  if (tid < 10) atomicAdd(&acc[plane * 10 + tid], wgAcc[tid]);
}

// Finalize: per-(b,c) closed-form KL, then per-batch sums over 13 classes
// computed as D = A x ones + C chains on V_WMMA_F32_16X16X4_F32 (one wave).
__global__ void BPKDV2_final_kernel(const float* __restrict__ acc,
                                    float* __restrict__ out)
{
  // q[quantity][batch][class(padded to 16)]: 0=klE, 1=klB, 2=cntE, 3=cntB
  __shared__ float q[4][8][16];
  const int t = threadIdx.x;
  for (int i = t; i < 4 * 8 * 16; i += 128) ((float*)q)[i] = 0.f;
  __syncthreads();

  if (t < NPLANE) {
    const float* a = acc + t * 10;
    const float cntE = a[0], AE = a[1], BE = a[2], CE = a[3], DE = a[4];
    const float cntB = a[5], AB = a[6], BB = a[7], CB = a[8], DB = a[9];
    if (cntE > 0.f) {                  // validity gate (edge count) gates BOTH
      const float Nf = (float)HW;
      const float rE = Nf - cntE;
      const float ZtE = AE + rE, ZsE = DE + rE;
      const float klE = (BE - CE) / ZtE + __logf(ZsE) - __logf(ZtE);
      const float rB = Nf - cntB;
      const float ZtB = AB + rB, ZsB = DB + rB;
      const float klB = (BB - CB) / ZtB + __logf(ZsB) - __logf(ZtB);
      const int b = t / 13, j = t % 13;
      q[0][b][j] = klE;
      q[1][b][j] = klB;
      q[2][b][j] = cntE;
      q[3][b][j] = cntB;
    }
  }
  __syncthreads();

  if (t < 32) {                        // wave 0, EXEC all ones
    const int lane = t;
    const int m  = lane & 15;          // A-matrix row (= batch, rows 8..15 zero)
    const int k0 = (lane < 16) ? 0 : 2;
    v2f ones; ones.x = 1.0f; ones.y = 1.0f;   // B = all-ones (layout-invariant)

    v8f d[4];
#pragma unroll
    for (int qq = 0; qq < 4; ++qq) {
      v8f cacc = {};
#pragma unroll
      for (int ch = 0; ch < 4; ++ch) {         // K chunks cover classes 0..15
        v2f amat;
        amat.x = (m < 8) ? q[qq][m][ch * 4 + k0]     : 0.0f;
        amat.y = (m < 8) ? q[qq][m][ch * 4 + k0 + 1] : 0.0f;
        cacc = __builtin_amdgcn_wmma_f32_16x16x4_f32(
            /*neg_a=*/false, amat, /*neg_b=*/false, ones,
            /*c_mod=*/(short)0, cacc, /*reuse_a=*/false, /*reuse_b=*/false);
      }
      d[qq] = cacc;                    // lane 0 element b == sum over classes, batch b
    }

    if (lane == 0) {
      float se = 0.f, sb = 0.f;
#pragma unroll
      for (int b = 0; b < 8; ++b) {
        const float klE = d[0][b], klB = d[1][b];
        const float ne  = d[2][b], nb  = d[3][b];
        se += (ne > 0.f) ? klE / ne : 0.f;
        sb += (nb > 0.f) ? klB / nb : 0.f;
      }
      out[0] = 500.0f * se / 8.0f;     // LOSS_WEIGHT * EDGE_WEIGHT * sum/B
      out[1] = 200.0f * sb / 8.0f;     // LOSS_WEIGHT * BODY_WEIGHT * sum/B
    }
  }
}

extern "C" void kernel_launch(void* const* d_in, const int* in_sizes, int n_in,
                              void* d_out, int out_size, void* d_ws, size_t ws_size,
                              hipStream_t stream) {
  const float* S = (const float*)d_in[0];   // preds_S (8,14,512,512) f32
  const float* T = (const float*)d_in[1];   // preds_T (8,14,512,512) f32
  const int*   L = (const int*)d_in[2];     // gt_labels (8,1,512,512) i32
  float* out = (float*)d_out;               // (loss_edges, loss_bodies)
  float* acc = (float*)d_ws;                // 104 planes * 10 sums

  BPKDV2_zero_kernel<<<(ACCN + 255) / 256, 256, 0, stream>>>(acc, ACCN);
  BPKDV2_main_kernel<<<NPLANE * 64, 256, 0, stream>>>(S, T, L, acc);
  BPKDV2_final_kernel<<<1, 128, 0, stream>>>(acc, out);
}